// DiscreteMamba2_9225589752175
// MI455X (gfx1250) — compile-verified
//
#include <hip/hip_runtime.h>

typedef __attribute__((ext_vector_type(2))) float v2f;
typedef __attribute__((ext_vector_type(8))) float v8f;

#define D_MODEL 2048
#define D_INNER 2048
#define NH      32
#define D_STATE 64
#define CONV_DIM 6144
#define IN_DIM  8224
#define B_SZ    2
#define L_SEQ   1024
#define ROWS    (B_SZ * L_SEQ)   // 2048
#define QCHUNK  64
#define NCHUNK  (L_SEQ / QCHUNK) // 16

__device__ __forceinline__ v8f wmma_f32(v2f a, v2f b, v8f c) {
  // D = A(16x4) * B(4x16) + C, full fp32
  return __builtin_amdgcn_wmma_f32_16x16x4_f32(false, a, false, b, (short)0, c,
                                               false, false);
}

// Raw 32-bit LDS byte offset for async-to-LDS VDST operand (addrspace(3)).
__device__ __forceinline__ unsigned lds_off(const void* p) {
  return (unsigned)(unsigned long long)(const __attribute__((address_space(3))) char*)p;
}

// One lane: async copy 16 bytes global -> LDS (ASYNCcnt-tracked, no VGPR data).
__device__ __forceinline__ void async_b128(unsigned lds_byte, unsigned goff_byte,
                                           const void* sbase) {
  asm volatile("global_load_async_to_lds_b128 %0, %1, %2"
               :: "v"(lds_byte), "v"(goff_byte), "s"(sbase) : "memory");
}
__device__ __forceinline__ void wait_async0() {
  asm volatile("s_wait_asynccnt 0x0" ::: "memory");
}

// ---------------------------------------------------------------------------
// C[M,N] = A[M,K] @ Bm[N,K]^T   (fp32 WMMA, 64x64 block tile, 8 waves,
// async global->LDS staging)
// ---------------------------------------------------------------------------
__global__ __launch_bounds__(256) void gemm_nt_wmma(
    const float* __restrict__ A, const float* __restrict__ Bm,
    float* __restrict__ C, int M, int N, int K) {
  __shared__ float As[64][36];   // pad 36: rows stay 16B-aligned, conflict-free frags
  __shared__ float BsT[64][36];  // [n][k]

  const int tid  = threadIdx.x;
  const int wv   = tid >> 5;
  const int lane = tid & 31;
  const int hf   = lane >> 4;    // wave32: which 16-lane half
  const int lr   = lane & 15;

  const int m0 = blockIdx.y * 64;
  const int n0 = blockIdx.x * 64;
  const int tm = (wv & 3) * 16;        // M sub-tile
  const int tn = (wv >> 2) * 32;       // N sub-tile pair

  // Zero the N-tail rows of the B tile once; their async loads are predicated
  // off below, so they stay zero for the whole K loop.
  if (n0 + 64 > N) {
    for (int t = tid; t < 512; t += 256) {
      const int row = t >> 3, kq = (t & 7) * 4;
      if (n0 + row >= N) {
        BsT[row][kq] = 0.f; BsT[row][kq + 1] = 0.f;
        BsT[row][kq + 2] = 0.f; BsT[row][kq + 3] = 0.f;
      }
    }
  }

  v8f acc0 = {};
  v8f acc1 = {};

  for (int k0 = 0; k0 < K; k0 += 32) {
    __syncthreads();
    // async stage A(64x32) and B(64x32) tiles straight into LDS
    for (int t = tid; t < 512; t += 256) {
      const int row = t >> 3;
      const int kq  = (t & 7) * 4;
      async_b128(lds_off(&As[row][kq]),
                 (unsigned)(((m0 + row) * K + k0 + kq) * 4), A);
      const int n = n0 + row;
      if (n < N)
        async_b128(lds_off(&BsT[row][kq]),
                   (unsigned)((n * K + k0 + kq) * 4), Bm);
    }
    wait_async0();
    __syncthreads();
#pragma unroll
    for (int kk = 0; kk < 32; kk += 4) {
      const int kb = kk + 2 * hf;  // lanes0-15: k,k+1 ; lanes16-31: k+2,k+3
      v2f a, b0, b1;
      a.x  = As[tm + lr][kb];        a.y  = As[tm + lr][kb + 1];
      b0.x = BsT[tn + lr][kb];       b0.y = BsT[tn + lr][kb + 1];
      b1.x = BsT[tn + 16 + lr][kb];  b1.y = BsT[tn + 16 + lr][kb + 1];
      acc0 = wmma_f32(a, b0, acc0);
      acc1 = wmma_f32(a, b1, acc1);
    }
  }
  // C/D layout: VGPR v -> M = v + 8*half, N = lane&15
#pragma unroll
  for (int v = 0; v < 8; ++v) {
    const int gm  = m0 + tm + v + 8 * hf;
    const int gn0 = n0 + tn + lr;
    const int gn1 = gn0 + 16;
    if (gn0 < N) C[(size_t)gm * N + gn0] = acc0[v];
    if (gn1 < N) C[(size_t)gm * N + gn1] = acc1[v];
  }
}

// ---------------------------------------------------------------------------
// Depthwise causal conv1d over the xBC slice of proj (identity activation)
// ---------------------------------------------------------------------------
__global__ __launch_bounds__(256) void conv_kernel(
    const float* __restrict__ proj, const float* __restrict__ cw,
    const float* __restrict__ cb, float* __restrict__ xc) {
  const int i = blockIdx.x * blockDim.x + threadIdx.x;
  if (i >= ROWS * CONV_DIM) return;
  const int c   = i % CONV_DIM;
  const int row = i / CONV_DIM;          // b*1024 + l
  const int l   = row & (L_SEQ - 1);
  const int bb  = row >> 10;
  float s = cb[c];
#pragma unroll
  for (int j = 0; j < 4; ++j) {
    const int ls = l - 3 + j;
    if (ls >= 0)
      s += cw[c * 4 + j] * proj[(size_t)(bb * L_SEQ + ls) * IN_DIM + c];
  }
  xc[(size_t)row * CONV_DIM + c] = s;
}

// ---------------------------------------------------------------------------
// SSD chunked scan: one workgroup per (b,h); 64x64 fp32 state in LDS;
// three WMMA matmul phases per 64-length chunk; async chunk staging.
// ---------------------------------------------------------------------------
__global__ __launch_bounds__(256) void ssd_kernel(
    const float* __restrict__ proj, const float* __restrict__ xc,
    const float* __restrict__ Dvec, float* __restrict__ ybuf) {
  const int bh = blockIdx.x;     // 0..63
  const int bb = bh >> 5;
  const int h  = bh & 31;
  const int tid  = threadIdx.x;
  const int wv   = tid >> 5;
  const int lane = tid & 31;
  const int hf   = lane >> 4;
  const int lr   = lane & 15;

  __shared__ float Cs[QCHUNK][68];   // stride 68: rows 16B-aligned (272B)
  __shared__ float Bs[QCHUNK][68];
  __shared__ float Xs[QCHUNK][68];
  __shared__ float Ss[QCHUNK][68];   // state [n][p]
  __shared__ float Gs[QCHUNK][68];   // masked transfer matrix [l][s]
  __shared__ float av[QCHUNK];
  __shared__ float csum[QCHUNK];
  __shared__ float wl[QCHUNK];

  for (int i = tid; i < QCHUNK * 64; i += 256) Ss[i >> 6][i & 63] = 0.f;
  const float Dh = Dvec[h];

  for (int t = 0; t < NCHUNK; ++t) {
    __syncthreads();
    // ---- async-stage chunk tiles of x, B, C (16B per lane per op) ----
    for (int i = tid; i < QCHUNK * 16; i += 256) {
      const int l = i >> 4, q = (i & 15) * 4;
      const unsigned rowb =
          (unsigned)(((bb * L_SEQ + t * QCHUNK + l) * CONV_DIM) * 4);
      async_b128(lds_off(&Xs[l][q]), rowb + (h * D_STATE + q) * 4, xc);
      async_b128(lds_off(&Bs[l][q]), rowb + (D_INNER + h * D_STATE + q) * 4, xc);
      async_b128(lds_off(&Cs[l][q]),
                 rowb + (2 * D_INNER + h * D_STATE + q) * 4, xc);
    }
    if (tid < QCHUNK) {
      const int row = bb * L_SEQ + t * QCHUNK + tid;
      const float al = proj[(size_t)row * IN_DIM + (IN_DIM - NH) + h];
      const float sp = (al > 20.f) ? al : log1pf(expf(al));  // softplus
      av[tid] = -sp;
    }
    wait_async0();
    __syncthreads();
    if (tid == 0) {  // inclusive prefix sum of a over the chunk (64 adds)
      float s = 0.f;
      for (int i = 0; i < QCHUNK; ++i) { s += av[i]; csum[i] = s; }
    }
    __syncthreads();
    if (tid < QCHUNK) wl[tid] = expf(csum[QCHUNK - 1] - csum[tid]);

    // ---- G = C @ B^T, then mask+decay -> Gs ----
    for (int r = 0; r < 2; ++r) {
      const int ti = wv * 2 + r;
      const int tm = (ti >> 2) * 16, tn = (ti & 3) * 16;
      v8f acc = {};
#pragma unroll
      for (int kk = 0; kk < 64; kk += 4) {
        const int kb = kk + 2 * hf;
        v2f a, b;
        a.x = Cs[tm + lr][kb]; a.y = Cs[tm + lr][kb + 1];
        b.x = Bs[tn + lr][kb]; b.y = Bs[tn + lr][kb + 1];
        acc = wmma_f32(a, b, acc);
      }
      const int scol = tn + lr;
#pragma unroll
      for (int v = 0; v < 8; ++v) {
        const int lrow = tm + v + 8 * hf;
        Gs[lrow][scol] =
            (lrow >= scol) ? acc[v] * expf(csum[lrow] - csum[scol]) : 0.f;
      }
    }
    __syncthreads();

    // ---- Y = Gs @ X  +  (C * e^{csum_l}) @ S  +  D*x ;  write to global ----
    for (int r = 0; r < 2; ++r) {
      const int ti = wv * 2 + r;
      const int tm = (ti >> 2) * 16, tn = (ti & 3) * 16;
      v8f acc = {};
#pragma unroll
      for (int kk = 0; kk < 64; kk += 4) {
        const int kb = kk + 2 * hf;
        v2f a, b;
        a.x = Gs[tm + lr][kb];   a.y = Gs[tm + lr][kb + 1];
        b.x = Xs[kb][tn + lr];   b.y = Xs[kb + 1][tn + lr];
        acc = wmma_f32(a, b, acc);
      }
      const float ecl = expf(csum[tm + lr]);
#pragma unroll
      for (int kk = 0; kk < 64; kk += 4) {
        const int kb = kk + 2 * hf;
        v2f a, b;
        a.x = Cs[tm + lr][kb] * ecl; a.y = Cs[tm + lr][kb + 1] * ecl;
        b.x = Ss[kb][tn + lr];       b.y = Ss[kb + 1][tn + lr];
        acc = wmma_f32(a, b, acc);
      }
      const int pcol = tn + lr;
#pragma unroll
      for (int v = 0; v < 8; ++v) {
        const int lrow = tm + v + 8 * hf;
        const int grow = bb * L_SEQ + t * QCHUNK + lrow;
        ybuf[(size_t)grow * D_INNER + h * (D_INNER / NH) + pcol] =
            acc[v] + Dh * Xs[lrow][pcol];
      }
    }
    __syncthreads();  // all reads of Ss done before the update writes

    // ---- state: S = e^{sum a} * S + B^T diag(wl) X ----
    const float et = expf(csum[QCHUNK - 1]);
    for (int r = 0; r < 2; ++r) {
      const int ti = wv * 2 + r;
      const int tm = (ti >> 2) * 16, tn = (ti & 3) * 16;  // tm over n, tn over p
      const int pcol = tn + lr;
      v8f acc;
#pragma unroll
      for (int v = 0; v < 8; ++v) acc[v] = et * Ss[tm + v + 8 * hf][pcol];
#pragma unroll
      for (int kk = 0; kk < 64; kk += 4) {
        const int kb = kk + 2 * hf;
        v2f a, b;
        a.x = Bs[kb][tm + lr] * wl[kb];          // A[n][l] = B[l][n]*w[l]
        a.y = Bs[kb + 1][tm + lr] * wl[kb + 1];
        b.x = Xs[kb][pcol];  b.y = Xs[kb + 1][pcol];
        acc = wmma_f32(a, b, acc);
      }
#pragma unroll
      for (int v = 0; v < 8; ++v) Ss[tm + v + 8 * hf][pcol] = acc[v];
    }
  }
}

// ---------------------------------------------------------------------------
// g = y * silu(z + z_bias)
// ---------------------------------------------------------------------------
__global__ __launch_bounds__(256) void gate_kernel(
    const float* __restrict__ proj, const float* __restrict__ ybuf,
    const float* __restrict__ zb, float* __restrict__ gbuf) {
  const int i = blockIdx.x * blockDim.x + threadIdx.x;
  if (i >= ROWS * D_INNER) return;
  const int r = i >> 11, c = i & (D_INNER - 1);
  const float z = proj[(size_t)r * IN_DIM + CONV_DIM + c] + zb[c];
  const float s = z / (1.f + expf(-z));
  gbuf[i] = ybuf[i] * s;
}

extern "C" void kernel_launch(void* const* d_in, const int* in_sizes, int n_in,
                              void* d_out, int out_size, void* d_ws,
                              size_t ws_size, hipStream_t stream) {
  (void)in_sizes; (void)n_in; (void)out_size; (void)ws_size;
  const float* u      = (const float*)d_in[0];
  const float* W_in   = (const float*)d_in[1];
  const float* conv_w = (const float*)d_in[2];
  const float* conv_b = (const float*)d_in[3];
  const float* z_bias = (const float*)d_in[4];
  const float* Dv     = (const float*)d_in[5];
  const float* W_out  = (const float*)d_in[6];
  float* out = (float*)d_out;

  char* ws = (char*)d_ws;
  const size_t PROJ_BYTES = (size_t)ROWS * IN_DIM * sizeof(float);    // 67.4 MB
  const size_t XC_BYTES   = (size_t)ROWS * CONV_DIM * sizeof(float);  // 50.3 MB
  float* proj = (float*)ws;
  float* xc   = (float*)(ws + PROJ_BYTES);
  float* ybuf = (float*)(ws + PROJ_BYTES + XC_BYTES);                 // 16.8 MB
  float* gbuf = xc;  // xc dead after ssd_kernel -> reuse for gated activations

  // 1) in_proj: [2048,8224] = u[2048,2048] @ W_in^T
  gemm_nt_wmma<<<dim3((IN_DIM + 63) / 64, ROWS / 64), 256, 0, stream>>>(
      u, W_in, proj, ROWS, IN_DIM, D_MODEL);
  // 2) causal depthwise conv over xBC
  conv_kernel<<<(ROWS * CONV_DIM + 255) / 256, 256, 0, stream>>>(
      proj, conv_w, conv_b, xc);
  // 3) SSD chunked scan (one workgroup per (b,h))
  ssd_kernel<<<B_SZ * NH, 256, 0, stream>>>(proj, xc, Dv, ybuf);
  // 4) gating
  gate_kernel<<<(ROWS * D_INNER + 255) / 256, 256, 0, stream>>>(
      proj, ybuf, z_bias, gbuf);
  // 5) out_proj: [2048,2048] = g @ W_out^T
  gemm_nt_wmma<<<dim3(D_MODEL / 64, ROWS / 64), 256, 0, stream>>>(
      gbuf, W_out, out, ROWS, D_MODEL, D_INNER);
}